// GATRecommendationModel_82197084111153
// MI455X (gfx1250) — compile-verified
//
#include <hip/hip_runtime.h>
#include <math.h>

typedef float v2f __attribute__((ext_vector_type(2)));
typedef float v8f __attribute__((ext_vector_type(8)));

#define NEG_SLOPE 0.2f

// ---------------------------------------------------------------------------
// float atomic max via int/uint ordering trick (valid with -inf init)
// ---------------------------------------------------------------------------
__device__ __forceinline__ void atomicMaxF32(float* addr, float val) {
    if (val >= 0.0f)
        atomicMax((int*)addr, __float_as_int(val));
    else
        atomicMin((unsigned int*)addr, __float_as_uint(val));
}

__global__ void zero_kernel(float* __restrict__ p, int n) {
    int i = blockIdx.x * blockDim.x + threadIdx.x;
    if (i < n) p[i] = 0.0f;
}

// ---------------------------------------------------------------------------
// Generic fp32 WMMA GEMM: D[M,N] = A[M,K] @ B[K,N]  (+bias, +relu optional)
// One wave (32 threads) per 16x16 output tile; V_WMMA_F32_16X16X4_F32.
// K, LDB, LDD are compile-time so all B-loads / D-stores use immediate
// offsets from a single base pointer (no per-iteration 64-bit address math),
// and the MAC loop fully unrolls.
// Optional rowidx indirection on A rows (for emb[x] gather fusion).
// ---------------------------------------------------------------------------
template <int K, int LDB, int LDD>
__global__ void wmma_gemm_f32_kernel(const float* __restrict__ A,
                                     const float* __restrict__ B,
                                     float* __restrict__ D,
                                     const int* __restrict__ rowidx,
                                     const float* __restrict__ bias,
                                     int relu, int M, int Ntiles, int lda) {
    const int tile = blockIdx.x;
    const int mt = tile / Ntiles;
    const int nt = tile % Ntiles;
    const int lane = threadIdx.x & 31;
    const int l  = lane & 15;
    const int hi = lane >> 4;                // lane-half selects K offset +2

    int r = mt * 16 + l;                     // A row for this lane
    if (r > M - 1) r = M - 1;                // clamp (stores are guarded)
    const int ar = rowidx ? rowidx[r] : r;
    const float* Ap = A + (size_t)ar * lda + 2 * hi;   // 8B-aligned (lda even)
    const int col = nt * 16 + l;                       // B/D column
    const float* Bp = B + (size_t)(2 * hi) * LDB + col;

    v8f acc = {};
#pragma unroll
    for (int k = 0; k < K; k += 4) {
        v2f a = *(const v2f*)(Ap + k);       // global_load_b64, imm offset
        v2f b;
        b.x = Bp[k * LDB];                   // imm offsets (compile-time)
        b.y = Bp[k * LDB + LDB];
        acc = __builtin_amdgcn_wmma_f32_16x16x4_f32(
                  false, a, false, b, (short)0, acc, false, false);
    }

    const float bv = bias ? bias[col] : 0.0f;
    float* Dp = D + (size_t)(mt * 16 + 8 * hi) * LDD + col;
    const int mbase = mt * 16 + 8 * hi;
#pragma unroll
    for (int i = 0; i < 8; ++i) {
        if (mbase + i < M) {
            float v = acc[i] + bv;
            if (relu) v = v > 0.0f ? v : 0.0f;
            Dp[i * LDD] = v;                 // imm offsets (compile-time)
        }
    }
}

// ---------------------------------------------------------------------------
// Per-(node,head) attention coefficients:  as = <x[n,h,:], a_src[h,:]>,
// ad likewise. Also initializes segment-max (-inf) and denom (0).
// ---------------------------------------------------------------------------
__global__ void alpha_init_kernel(const float* __restrict__ X,
                                  const float* __restrict__ a_s,
                                  const float* __restrict__ a_d,
                                  float* __restrict__ as_o,
                                  float* __restrict__ ad_o,
                                  float* __restrict__ emax,
                                  float* __restrict__ den,
                                  int Nn, int H, int C) {
    int i = blockIdx.x * blockDim.x + threadIdx.x;
    if (i >= Nn * H) return;
    int n = i / H, h = i % H;
    const float* xp  = X   + (size_t)n * H * C + (size_t)h * C;
    const float* asp = a_s + (size_t)h * C;
    const float* adp = a_d + (size_t)h * C;
    float s = 0.0f, d = 0.0f;
    for (int c = 0; c < C; ++c) {
        float v = xp[c];
        s += v * asp[c];
        d += v * adp[c];
    }
    as_o[i] = s;
    ad_o[i] = d;
    emax[i] = -INFINITY;
    den[i]  = 0.0f;
}

__device__ __forceinline__ void edge_ends(const int* src, const int* dst,
                                          int e, int E, int& s, int& d) {
    if (e < E) { s = src[e]; d = dst[e]; }
    else       { s = e - E;  d = e - E; }   // appended self-loops
}

// segment max of leaky_relu(as[src]+ad[dst]) over incoming edges of dst
__global__ void edge_max_kernel(const int* __restrict__ src,
                                const int* __restrict__ dst,
                                const float* __restrict__ as,
                                const float* __restrict__ ad,
                                float* __restrict__ emax,
                                int E, int Nn, int H) {
    int i = blockIdx.x * blockDim.x + threadIdx.x;
    int tot = (E + Nn) * H;
    if (i >= tot) return;
    int e = i / H, h = i % H;
    int s, d; edge_ends(src, dst, e, E, s, d);
    float v = as[s * H + h] + ad[d * H + h];
    v = v > 0.0f ? v : NEG_SLOPE * v;
    atomicMaxF32(&emax[d * H + h], v);
}

// e_exp per edge + segment sum of exps
__global__ void edge_exp_kernel(const int* __restrict__ src,
                                const int* __restrict__ dst,
                                const float* __restrict__ as,
                                const float* __restrict__ ad,
                                const float* __restrict__ emax,
                                float* __restrict__ den,
                                float* __restrict__ eexp,
                                int E, int Nn, int H) {
    int i = blockIdx.x * blockDim.x + threadIdx.x;
    int tot = (E + Nn) * H;
    if (i >= tot) return;
    int e = i / H, h = i % H;
    int s, d; edge_ends(src, dst, e, E, s, d);
    float v = as[s * H + h] + ad[d * H + h];
    v = v > 0.0f ? v : NEG_SLOPE * v;
    float ex = expf(v - emax[d * H + h]);
    eexp[i] = ex;
    atomicAdd(&den[d * H + h], ex);
}

// out[dst] += x[src] * alpha   (one block per edge, H*C threads)
__global__ void edge_agg_kernel(const int* __restrict__ src,
                                const int* __restrict__ dst,
                                const float* __restrict__ X,
                                const float* __restrict__ eexp,
                                const float* __restrict__ den,
                                float* __restrict__ out,
                                int E, int Nn, int H, int C) {
    int e = blockIdx.x;
    int t = threadIdx.x;                 // 0 .. H*C-1
    int h = t / C;
    int s, d; edge_ends(src, dst, e, E, s, d);
    float alpha = eexp[(size_t)e * H + h] / (den[d * H + h] + 1e-16f);
    int HC = H * C;
    atomicAdd(&out[(size_t)d * HC + t], X[(size_t)s * HC + t] * alpha);
}

// X += bias (per feature); mode 1 applies ELU afterwards
__global__ void bias_act_kernel(float* __restrict__ X,
                                const float* __restrict__ bias,
                                int n, int F, int elu) {
    int i = blockIdx.x * blockDim.x + threadIdx.x;
    if (i >= n) return;
    int f = i % F;
    float v = X[i] + bias[f];
    if (elu) v = v > 0.0f ? v : expm1f(v);
    X[i] = v;
}

// combined[b] = [h2[user[b]], h2[item[b]]]   (64 + 64 = 128)
__global__ void gather_concat_kernel(const float* __restrict__ h2,
                                     const int* __restrict__ ui,
                                     const int* __restrict__ ii,
                                     float* __restrict__ comb, int Bn) {
    int i = blockIdx.x * blockDim.x + threadIdx.x;
    if (i >= Bn * 128) return;
    int b = i >> 7, c = i & 127;
    comb[i] = (c < 64) ? h2[(size_t)ui[b] * 64 + c]
                       : h2[(size_t)ii[b] * 64 + (c - 64)];
}

// pred = sigmoid(hid @ Wp2 + bp2); one wave per batch row, shuffle reduce
__global__ void mlp_out_kernel(const float* __restrict__ hid,
                               const float* __restrict__ w,
                               const float* __restrict__ b,
                               float* __restrict__ out, int Bn) {
    int gid = blockIdx.x * blockDim.x + threadIdx.x;
    int row = gid >> 5, lane = gid & 31;
    if (row >= Bn) return;
    float s = 0.0f;
#pragma unroll
    for (int c = lane; c < 128; c += 32) s += hid[(size_t)row * 128 + c] * w[c];
    for (int off = 16; off; off >>= 1) s += __shfl_down(s, off, 32);
    if (lane == 0) out[row] = 1.0f / (1.0f + expf(-(s + b[0])));
}

// ---------------------------------------------------------------------------
extern "C" void kernel_launch(void* const* d_in, const int* in_sizes, int n_in,
                              void* d_out, int out_size, void* d_ws, size_t ws_size,
                              hipStream_t stream) {
    const int*   x      = (const int*)d_in[0];
    const int*   eidx   = (const int*)d_in[1];
    const int*   ui     = (const int*)d_in[2];
    const int*   ii     = (const int*)d_in[3];
    const float* emb    = (const float*)d_in[4];
    const float* W1     = (const float*)d_in[5];
    const float* a_src1 = (const float*)d_in[6];
    const float* a_dst1 = (const float*)d_in[7];
    const float* b1     = (const float*)d_in[8];
    const float* W2     = (const float*)d_in[9];
    const float* a_src2 = (const float*)d_in[10];
    const float* a_dst2 = (const float*)d_in[11];
    const float* b2     = (const float*)d_in[12];
    const float* Wp1    = (const float*)d_in[13];
    const float* bp1    = (const float*)d_in[14];
    const float* Wp2    = (const float*)d_in[15];
    const float* bp2    = (const float*)d_in[16];

    const int Nn   = in_sizes[0];
    const int E    = in_sizes[1] / 2;
    const int Bn   = in_sizes[2];
    const int Etot = E + Nn;
    const int* src = eidx;
    const int* dst = eidx + E;
    float* outp = (float*)d_out;

    // ---- workspace carve-up (floats) ----
    float* p = (float*)d_ws;
    float* x1    = p; p += (size_t)Nn * 128;   // layer1 features; reused as h2pre
    float* out1  = p; p += (size_t)Nn * 128;   // layer1 aggregate -> h1 (in place)
    float* as1   = p; p += (size_t)Nn * 4;
    float* ad1   = p; p += (size_t)Nn * 4;
    float* emax1 = p; p += (size_t)Nn * 4;
    float* den1  = p; p += (size_t)Nn * 4;
    float* eexp  = p; p += (size_t)Etot * 4;   // reused for layer2 (needs Etot)
    float* out2  = p; p += (size_t)Nn * 64;    // layer2 aggregate -> h2 (in place)
    float* as2   = p; p += (size_t)Nn;
    float* ad2   = p; p += (size_t)Nn;
    float* emax2 = p; p += (size_t)Nn;
    float* den2  = p; p += (size_t)Nn;
    float* comb  = p; p += (size_t)Bn * 128;
    float* hid   = p; p += (size_t)Bn * 128;
    float* h2pre = x1;                         // alias: x1 dead after layer-1 agg

    const int T = 256;
    auto cdiv = [](int a, int b) { return (a + b - 1) / b; };

    // zero the scatter accumulators
    zero_kernel<<<cdiv(Nn * 128, T), T, 0, stream>>>(out1, Nn * 128);
    zero_kernel<<<cdiv(Nn * 64,  T), T, 0, stream>>>(out2, Nn * 64);

    // ---- layer 1: x1 = emb[x] @ W1  (heads=4, ch=32 -> N=128, K=64) ----
    {
        int grid = cdiv(Nn, 16) * (128 / 16);
        wmma_gemm_f32_kernel<64, 128, 128><<<grid, 32, 0, stream>>>(
            emb, W1, x1, x, nullptr, 0, Nn, 128 / 16, 64);
    }
    alpha_init_kernel<<<cdiv(Nn * 4, T), T, 0, stream>>>(
        x1, a_src1, a_dst1, as1, ad1, emax1, den1, Nn, 4, 32);
    edge_max_kernel<<<cdiv(Etot * 4, T), T, 0, stream>>>(
        src, dst, as1, ad1, emax1, E, Nn, 4);
    edge_exp_kernel<<<cdiv(Etot * 4, T), T, 0, stream>>>(
        src, dst, as1, ad1, emax1, den1, eexp, E, Nn, 4);
    edge_agg_kernel<<<Etot, 128, 0, stream>>>(
        src, dst, x1, eexp, den1, out1, E, Nn, 4, 32);
    bias_act_kernel<<<cdiv(Nn * 128, T), T, 0, stream>>>(out1, b1, Nn * 128, 128, 1);

    // ---- layer 2: h2pre = h1 @ W2  (heads=1, ch=64 -> N=64, K=128) ----
    {
        int grid = cdiv(Nn, 16) * (64 / 16);
        wmma_gemm_f32_kernel<128, 64, 64><<<grid, 32, 0, stream>>>(
            out1, W2, h2pre, nullptr, nullptr, 0, Nn, 64 / 16, 128);
    }
    alpha_init_kernel<<<cdiv(Nn, T), T, 0, stream>>>(
        h2pre, a_src2, a_dst2, as2, ad2, emax2, den2, Nn, 1, 64);
    edge_max_kernel<<<cdiv(Etot, T), T, 0, stream>>>(
        src, dst, as2, ad2, emax2, E, Nn, 1);
    edge_exp_kernel<<<cdiv(Etot, T), T, 0, stream>>>(
        src, dst, as2, ad2, emax2, den2, eexp, E, Nn, 1);
    edge_agg_kernel<<<Etot, 64, 0, stream>>>(
        src, dst, h2pre, eexp, den2, out2, E, Nn, 1, 64);
    bias_act_kernel<<<cdiv(Nn * 64, T), T, 0, stream>>>(out2, b2, Nn * 64, 64, 0);

    // ---- MLP head ----
    gather_concat_kernel<<<cdiv(Bn * 128, T), T, 0, stream>>>(out2, ui, ii, comb, Bn);
    {
        int grid = cdiv(Bn, 16) * (128 / 16);
        wmma_gemm_f32_kernel<128, 128, 128><<<grid, 32, 0, stream>>>(
            comb, Wp1, hid, nullptr, bp1, 1, Bn, 128 / 16, 128);
    }
    mlp_out_kernel<<<cdiv(Bn * 32, T), T, 0, stream>>>(hid, Wp2, bp2, outp, Bn);
}